// EncodeProcessDecode_73194832659230
// MI455X (gfx1250) — compile-verified
//
#include <hip/hip_runtime.h>

#define NN 50000
#define EE 800000
#define DD 128
#define ETOT (EE + NN)
#define SLOPE 0.2f

typedef float v2f __attribute__((ext_vector_type(2)));
typedef float v8f __attribute__((ext_vector_type(8)));

// ---------------------------------------------------------------------------
// GEMM: H[N,128] (+)= A[N,128] @ W[128,128] using V_WMMA_F32_16X16X4_F32.
// One wave computes one 16x16 tile; 8 waves/block cover all 128 output cols.
// Grid.x = N/16 = 3125 exactly -> no partial tiles, EXEC all-1s for WMMA.
// Fragment layouts per CDNA5 ISA 7.12.2 (32-bit A 16x4, 32-bit C/D 16x16).
// ---------------------------------------------------------------------------
__global__ __launch_bounds__(256) void gemm_wmma_f32(
    const float* __restrict__ A, const float* __restrict__ W,
    float* __restrict__ H, int accumulate) {
  const int lane = threadIdx.x & 31;
  const int wv   = threadIdx.x >> 5;
  const int m0   = blockIdx.x * 16;
  const int n0   = wv * 16;
  const int half = lane >> 4;   // 0: lanes 0-15, 1: lanes 16-31
  const int l15  = lane & 15;
  const int row  = m0 + l15;    // A-matrix row held by this lane
  const int col  = n0 + l15;    // B/C/D column held by this lane

  v8f acc;
  if (accumulate) {
#pragma unroll
    for (int r = 0; r < 8; ++r)
      acc[r] = H[(size_t)(m0 + half * 8 + r) * DD + col];
  } else {
#pragma unroll
    for (int r = 0; r < 8; ++r) acc[r] = 0.0f;
  }

  const float* arow = A + (size_t)row * DD;
#pragma unroll 4
  for (int k = 0; k < DD; k += 4) {
    v2f av, bv;
    // A 16x4 f32: lanes 0-15 hold K = k,k+1 ; lanes 16-31 hold K = k+2,k+3
    av[0] = arow[k + 2 * half];
    av[1] = arow[k + 2 * half + 1];
    // B 4x16 f32: mirrored striping across lanes within the VGPR pair
    bv[0] = W[(size_t)(k + 2 * half) * DD + col];
    bv[1] = W[(size_t)(k + 2 * half + 1) * DD + col];
    acc = __builtin_amdgcn_wmma_f32_16x16x4_f32(false, av, false, bv,
                                                (short)0, acc, false, false);
  }
#pragma unroll
  for (int r = 0; r < 8; ++r)
    H[(size_t)(m0 + half * 8 + r) * DD + col] = acc[r];
}

// ---------------------------------------------------------------------------
// Per-node attention scores s_src = h.a_src, s_dst = h.a_dst (one wave/node),
// fused with init of segment-max (ordered-uint 0 == -inf) and denom.
// ---------------------------------------------------------------------------
__global__ __launch_bounds__(256) void scores_init(
    const float* __restrict__ H, const float* __restrict__ a_s,
    const float* __restrict__ a_d, float* __restrict__ s_src,
    float* __restrict__ s_dst, unsigned* __restrict__ m_ord,
    float* __restrict__ denom) {
  const int lane = threadIdx.x & 31;
  const int node = blockIdx.x * 8 + (threadIdx.x >> 5);
  if (node >= NN) return;
  const float4 hv = *(const float4*)(H + (size_t)node * DD + lane * 4);
  const float4 as = *(const float4*)(a_s + lane * 4);
  const float4 ad = *(const float4*)(a_d + lane * 4);
  float ps = hv.x * as.x + hv.y * as.y + hv.z * as.z + hv.w * as.w;
  float pd = hv.x * ad.x + hv.y * ad.y + hv.z * ad.z + hv.w * ad.w;
#pragma unroll
  for (int off = 16; off > 0; off >>= 1) {
    ps += __shfl_down(ps, off, 32);
    pd += __shfl_down(pd, off, 32);
  }
  if (lane == 0) {
    s_src[node] = ps;
    s_dst[node] = pd;
    m_ord[node] = 0u;      // minimum of the ordered-uint float encoding
    denom[node] = 0.0f;
  }
}

__device__ __forceinline__ unsigned f2ord(float f) {
  unsigned u = __float_as_uint(f);
  return (u & 0x80000000u) ? ~u : (u | 0x80000000u);
}
__device__ __forceinline__ float ord2f(unsigned o) {
  return (o & 0x80000000u) ? __uint_as_float(o & 0x7FFFFFFFu)
                           : __uint_as_float(~o);
}

// Edge pass 1: e = leaky_relu(s_src[src]+s_dst[dst]); segment max via atomicMax
__global__ __launch_bounds__(256) void edge_max(
    const int* __restrict__ ei, const float* __restrict__ s_src,
    const float* __restrict__ s_dst, float* __restrict__ wbuf,
    unsigned* __restrict__ m_ord) {
  int e = blockIdx.x * blockDim.x + threadIdx.x;
  if (e >= ETOT) return;
  int s, d;
  if (e < EE) { s = ei[e]; d = ei[EE + e]; } else { s = d = e - EE; }
  float v = s_src[s] + s_dst[d];
  v = v > 0.0f ? v : SLOPE * v;
  wbuf[e] = v;
  atomicMax(&m_ord[d], f2ord(v));
}

// Edge pass 2: w = exp(e - m[dst]); segment sum into denom
__global__ __launch_bounds__(256) void edge_exp(
    const int* __restrict__ ei, const unsigned* __restrict__ m_ord,
    float* __restrict__ wbuf, float* __restrict__ denom) {
  int e = blockIdx.x * blockDim.x + threadIdx.x;
  if (e >= ETOT) return;
  int d = (e < EE) ? ei[EE + e] : (e - EE);
  float z = __expf(wbuf[e] - ord2f(m_ord[d]));
  wbuf[e] = z;
  atomicAdd(&denom[d], z);
}

// out[n,:] = bias (softmax-weighted messages are accumulated on top)
__global__ __launch_bounds__(256) void init_out(float* __restrict__ out,
                                                const float* __restrict__ b) {
  int i = blockIdx.x * blockDim.x + threadIdx.x;  // N*D / 256 exact
  out[i] = b[i & (DD - 1)];
}

// Edge pass 3: one wave per edge; lane l moves features 4l..4l+3.
// b128 gather from h[src] (L2-resident) + 4x global_atomic_add_f32 scatter.
__global__ __launch_bounds__(256) void edge_scatter(
    const int* __restrict__ ei, const float* __restrict__ H,
    const float* __restrict__ wbuf, const float* __restrict__ denom,
    float* __restrict__ out) {
  int e    = (int)((blockIdx.x * blockDim.x + threadIdx.x) >> 5);
  int lane = threadIdx.x & 31;
  if (e >= ETOT) return;
  int s, d;
  if (e < EE) { s = ei[e]; d = ei[EE + e]; } else { s = d = e - EE; }
  float alpha = wbuf[e] / denom[d];
  const float4 hv = *(const float4*)(H + (size_t)s * DD + lane * 4);
  float* o = out + (size_t)d * DD + lane * 4;
  atomicAdd(o + 0, alpha * hv.x);
  atomicAdd(o + 1, alpha * hv.y);
  atomicAdd(o + 2, alpha * hv.z);
  atomicAdd(o + 3, alpha * hv.w);
}

// ---------------------------------------------------------------------------
// One full GAT layer. A2 == nullptr -> K=128; else h = A1@W[0:128] + A2@W[128:256]
// ---------------------------------------------------------------------------
static void run_gat(const float* A1, const float* A2, const float* Wmat,
                    const float* a_s, const float* a_d, const float* bias,
                    const int* ei, float* ht, float* out, float* s_src,
                    float* s_dst, unsigned* m_ord, float* denom, float* wbuf,
                    hipStream_t stream) {
  gemm_wmma_f32<<<NN / 16, 256, 0, stream>>>(A1, Wmat, ht, 0);
  if (A2)
    gemm_wmma_f32<<<NN / 16, 256, 0, stream>>>(A2, Wmat + DD * DD, ht, 1);
  scores_init<<<NN / 8, 256, 0, stream>>>(ht, a_s, a_d, s_src, s_dst, m_ord,
                                          denom);
  edge_max<<<(ETOT + 255) / 256, 256, 0, stream>>>(ei, s_src, s_dst, wbuf,
                                                   m_ord);
  edge_exp<<<(ETOT + 255) / 256, 256, 0, stream>>>(ei, m_ord, wbuf, denom);
  init_out<<<(NN * DD) / 256, 256, 0, stream>>>(out, bias);
  edge_scatter<<<(ETOT + 7) / 8, 256, 0, stream>>>(ei, ht, wbuf, denom, out);
}

extern "C" void kernel_launch(void* const* d_in, const int* in_sizes, int n_in,
                              void* d_out, int out_size, void* d_ws,
                              size_t ws_size, hipStream_t stream) {
  const float* x    = (const float*)d_in[0];
  const int*   ei   = (const int*)d_in[1];
  const float* We   = (const float*)d_in[2];
  const float* ae_s = (const float*)d_in[3];
  const float* ae_d = (const float*)d_in[4];
  const float* be   = (const float*)d_in[5];
  const float* Wp   = (const float*)d_in[6];
  const float* ap_s = (const float*)d_in[7];
  const float* ap_d = (const float*)d_in[8];
  const float* bp   = (const float*)d_in[9];
  const float* Wd   = (const float*)d_in[10];
  const float* ad_s = (const float*)d_in[11];
  const float* ad_d = (const float*)d_in[12];
  const float* bd   = (const float*)d_in[13];
  float*       out  = (float*)d_out;

  const size_t ND = (size_t)NN * DD;
  float*    enc   = (float*)d_ws;
  float*    hA    = enc + ND;
  float*    hB    = hA + ND;
  float*    ht    = hB + ND;
  float*    s_src = ht + ND;
  float*    s_dst = s_src + NN;
  unsigned* m_ord = (unsigned*)(s_dst + NN);
  float*    denom = (float*)(m_ord + NN);
  float*    wbuf  = denom + NN;

  // encode: encoded = GAT(x; We)
  run_gat(x, nullptr, We, ae_s, ae_d, be, ei, ht, enc, s_src, s_dst, m_ord,
          denom, wbuf, stream);
  // processor step 1: hA = GAT([enc, enc]; Wp)
  run_gat(enc, enc, Wp, ap_s, ap_d, bp, ei, ht, hA, s_src, s_dst, m_ord,
          denom, wbuf, stream);
  // processor step 2: hB = GAT([hA, enc]; Wp)
  run_gat(hA, enc, Wp, ap_s, ap_d, bp, ei, ht, hB, s_src, s_dst, m_ord,
          denom, wbuf, stream);
  // decode: out = GAT(hB; Wd)
  run_gat(hB, nullptr, Wd, ad_s, ad_d, bd, ei, ht, out, s_src, s_dst, m_ord,
          denom, wbuf, stream);
}